// HGT_83056077570928
// MI455X (gfx1250) — compile-verified
//
#include <hip/hip_runtime.h>
#include <hip/hip_bf16.h>

// ---------------------------------------------------------------------------
// HGT fund->customer SAGE layer, fused for MI455X (gfx1250, wave32, WMMA).
//
// Pipeline (all on `stream`):
//   0) memset summed[500000x64] + cnt[500000] (contiguous in ws)
//   1) fold_kernel: Wl2 = W_l@W_out, Wr2 = W_r@W_out, b2 = b_l@W_out + b_out
//   2) hf_kernel:   h_f = relu(x_fund * W_fund + b_fund)        [50000x64]
//   3) scatter:     atomic mean-aggregation of h_f over 4M edges
//   4) cust_kernel: per 16-row tile, WMMA f32 16x16x4 GEMM
//                   h_c = relu(X[16x101] @ W_cust[101x64] + b_cust),
//                   then out = mean@Wl2 + h_c@Wr2 + b2 via VALU reduction.
// ---------------------------------------------------------------------------

#define N_CUST 500000
#define N_FUND 50000
#define N_EDGE 4000000
#define D_CUST 101
#define HID    64
#define KPAD   106            // padded K (zeros for k>=101); 106 dwords/row -> no LDS bank conflicts, 8B aligned
#define NTILES (N_CUST / 16)  // 31250

typedef __attribute__((ext_vector_type(2))) float v2f;
typedef __attribute__((ext_vector_type(8))) float v8f;

// ---- 1) fold the output head: Wl2/Wr2 [64][2], b2 [2] ----------------------
__global__ __launch_bounds__(128) void fold_kernel(
    const float* __restrict__ Wl, const float* __restrict__ bl,
    const float* __restrict__ Wr, const float* __restrict__ Wo,
    const float* __restrict__ bo,
    float* __restrict__ Wl2, float* __restrict__ Wr2, float* __restrict__ b2) {
  int t = threadIdx.x;           // 0..127
  int c = t >> 1, o = t & 1;     // c: hidden col, o: output idx
  float sl = 0.f, sr = 0.f;
  for (int j = 0; j < HID; ++j) {
    float w = Wo[j * 2 + o];
    sl += Wl[c * HID + j] * w;
    sr += Wr[c * HID + j] * w;
  }
  Wl2[c * 2 + o] = sl;
  Wr2[c * 2 + o] = sr;
  if (c == 0) {
    float s = bo[o];
    for (int j = 0; j < HID; ++j) s += bl[j] * Wo[j * 2 + o];
    b2[o] = s;
  }
}

// ---- 2) h_f = relu(x_fund * W_fund + b_fund) -------------------------------
__global__ __launch_bounds__(256) void hf_kernel(
    const float* __restrict__ xf, const float* __restrict__ Wf,
    const float* __restrict__ bf, float* __restrict__ hf) {
  int tid = blockIdx.x * 256 + threadIdx.x;   // over N_FUND*HID
  int i = tid >> 6, j = tid & 63;
  float h = xf[i] * Wf[j] + bf[j];
  hf[tid] = fmaxf(h, 0.0f);
}

// ---- 3) edge scatter: one wave per edge ------------------------------------
__global__ __launch_bounds__(256) void scatter_kernel(
    const int* __restrict__ src, const int* __restrict__ dst,
    const float* __restrict__ hf, float* __restrict__ summed,
    float* __restrict__ cnt) {
  int e = blockIdx.x * 8 + (threadIdx.x >> 5);      // 8 waves/block, 1 edge/wave
  int l = threadIdx.x & 31;
  int s = src[e];
  int d = dst[e];
  float v0 = hf[(size_t)s * HID + l];
  float v1 = hf[(size_t)s * HID + 32 + l];
  atomicAdd(&summed[(size_t)d * HID + l], v0);
  atomicAdd(&summed[(size_t)d * HID + 32 + l], v1);
  if (l == 0) atomicAdd(&cnt[d], 1.0f);
}

// ---- 4) fused customer kernel: WMMA GEMM + mean + folded head --------------
__global__ __launch_bounds__(256) void cust_kernel(
    const float* __restrict__ x, const float* __restrict__ Wc,
    const float* __restrict__ bc, const float* __restrict__ summed,
    const float* __restrict__ cnt, const float* __restrict__ Wl2,
    const float* __restrict__ Wr2, const float* __restrict__ b2,
    float* __restrict__ out, int ntiles) {
  __shared__ float Wt[HID][KPAD];   // W_cust transposed + zero-padded (26.5 KB)

  for (int idx = threadIdx.x; idx < HID * KPAD; idx += 256) {
    int col = idx / KPAD, k = idx % KPAD;
    Wt[col][k] = (k < D_CUST) ? Wc[k * HID + col] : 0.0f;
  }
  __syncthreads();

  int wave = threadIdx.x >> 5;
  int lane = threadIdx.x & 31;
  int tile = blockIdx.x * 8 + wave;
  if (tile >= ntiles) return;       // wave-uniform; EXEC stays all-ones for WMMA

  int base = tile * 16;
  int row  = lane & 15;             // A: M row; B/C/D: N col within 16-tile
  int hl   = lane >> 4;             // half: A/B K offset +2, C/D M offset +8
  const float* xrow = x + (size_t)(base + row) * D_CUST;

  v8f acc[4];
  #pragma unroll
  for (int nt = 0; nt < 4; ++nt) acc[nt] = (v8f){};

  // main K loop: k = 0..96 covers K rows 0..99
  for (int k = 0; k < 100; k += 4) {
    int kk = k + 2 * hl;
    v2f a; a.x = xrow[kk]; a.y = xrow[kk + 1];
    #pragma unroll
    for (int nt = 0; nt < 4; ++nt) {
      v2f b = *(const v2f*)&Wt[nt * 16 + row][kk];
      acc[nt] = __builtin_amdgcn_wmma_f32_16x16x4_f32(
          false, a, false, b, (short)0, acc[nt], false, false);
    }
  }
  // tail step: K row 100 only (B side is zero-padded; A forced to 0 past 100)
  {
    int kk = 100 + 2 * hl;
    v2f a; a.x = (hl == 0) ? xrow[100] : 0.0f; a.y = 0.0f;
    #pragma unroll
    for (int nt = 0; nt < 4; ++nt) {
      v2f b = *(const v2f*)&Wt[nt * 16 + row][kk];
      acc[nt] = __builtin_amdgcn_wmma_f32_16x16x4_f32(
          false, a, false, b, (short)0, acc[nt], false, false);
    }
  }

  // folded head weights for this lane's 4 columns (col = nt*16 + row)
  float wl0[4], wl1[4], wr0[4], wr1[4], bcv[4];
  #pragma unroll
  for (int nt = 0; nt < 4; ++nt) {
    int col = nt * 16 + row;
    wl0[nt] = Wl2[col * 2 + 0];
    wl1[nt] = Wl2[col * 2 + 1];
    wr0[nt] = Wr2[col * 2 + 0];
    wr1[nt] = Wr2[col * 2 + 1];
    bcv[nt] = bc[col];
  }
  float b20 = b2[0], b21 = b2[1];

  float invc[8];
  #pragma unroll
  for (int v = 0; v < 8; ++v) {
    float c = cnt[base + v + 8 * hl];
    invc[v] = 1.0f / fmaxf(c, 1.0f);
  }

  // per output row m = v + 8*hl: dot over 64 cols (4 per lane x 16 lanes/half)
  #pragma unroll
  for (int v = 0; v < 8; ++v) {
    int m = v + 8 * hl;
    float p0 = 0.f, p1 = 0.f;
    #pragma unroll
    for (int nt = 0; nt < 4; ++nt) {
      float hc = fmaxf(acc[nt][v] + bcv[nt], 0.0f);
      float mn = summed[(size_t)(base + m) * HID + nt * 16 + row] * invc[v];
      p0 += hc * wr0[nt] + mn * wl0[nt];
      p1 += hc * wr1[nt] + mn * wl1[nt];
    }
    // reduce across the 16 lanes of this half (xor stays within half)
    #pragma unroll
    for (int off = 1; off < 16; off <<= 1) {
      p0 += __shfl_xor(p0, off, 32);
      p1 += __shfl_xor(p1, off, 32);
    }
    if (row == v) {
      float2 o2 = make_float2(p0 + b20, p1 + b21);
      *(float2*)&out[(size_t)(base + m) * 2] = o2;
    }
  }
}

// ---------------------------------------------------------------------------
extern "C" void kernel_launch(void* const* d_in, const int* in_sizes, int n_in,
                              void* d_out, int out_size, void* d_ws, size_t ws_size,
                              hipStream_t stream) {
  const float* x_cust = (const float*)d_in[0];
  const float* x_fund = (const float*)d_in[1];
  const int*   src    = (const int*)d_in[2];   // jax x64-off: int32 on device
  const int*   dst    = (const int*)d_in[3];
  const float* Wc     = (const float*)d_in[4];
  const float* bcst   = (const float*)d_in[5];
  const float* Wf     = (const float*)d_in[6];
  const float* bf     = (const float*)d_in[7];
  const float* Wl     = (const float*)d_in[8];
  const float* bl     = (const float*)d_in[9];
  const float* Wr     = (const float*)d_in[10];
  const float* Wo     = (const float*)d_in[11];
  const float* bo     = (const float*)d_in[12];
  float* out = (float*)d_out;

  // workspace layout (floats): hf | summed | cnt | Wl2 | Wr2 | b2  (~143 MB)
  float* ws     = (float*)d_ws;
  float* hf     = ws;
  float* summed = hf + (size_t)N_FUND * HID;
  float* cnt    = summed + (size_t)N_CUST * HID;
  float* Wl2    = cnt + N_CUST;
  float* Wr2    = Wl2 + HID * 2;
  float* b2     = Wr2 + HID * 2;

  // zero accumulators (summed + cnt are contiguous)
  hipMemsetAsync(summed, 0, ((size_t)N_CUST * HID + N_CUST) * sizeof(float), stream);

  fold_kernel<<<1, 128, 0, stream>>>(Wl, bl, Wr, Wo, bo, Wl2, Wr2, b2);
  hf_kernel<<<(N_FUND * HID) / 256, 256, 0, stream>>>(x_fund, Wf, bf, hf);
  scatter_kernel<<<N_EDGE / 8, 256, 0, stream>>>(src, dst, hf, summed, cnt);
  cust_kernel<<<(NTILES + 7) / 8, 256, 0, stream>>>(
      x_cust, Wc, bcst, summed, cnt, Wl2, Wr2, b2, out, NTILES);
}